// FDN_41764261986379
// MI455X (gfx1250) — compile-verified
//
#include <hip/hip_runtime.h>
#include <hip/hip_bf16.h>
#include <math.h>

// Problem constants (match reference)
#define NUM_CHANNEL 16
#define NUM_SAMPLE  16000
#define NUM_FREQ    8001      // rfft bins
#define BATCH       32
#define N1          1000      // long DFT length  (16000 = 16 * 1000)
#define N1P         1008      // N1 padded to multiple of 16
#define N2          16        // short DFT length
#define NJ          512       // BATCH * N2 columns of the final GEMM

#define TWO_PI      6.283185307179586f

typedef __attribute__((ext_vector_type(2))) float v2f;
typedef __attribute__((ext_vector_type(8))) float v8f;

// 16th roots of unity: cos/sin(2*pi*p/16), p = 0..15 (exact constants)
__constant__ float COS16[16] = {
     1.0f,  0.9238795325f,  0.7071067812f,  0.3826834324f,
     0.0f, -0.3826834324f, -0.7071067812f, -0.9238795325f,
    -1.0f, -0.9238795325f, -0.7071067812f, -0.3826834324f,
     0.0f,  0.3826834324f,  0.7071067812f,  0.9238795325f };
__constant__ float SIN16[16] = {
     0.0f,  0.3826834324f,  0.7071067812f,  0.9238795325f,
     1.0f,  0.9238795325f,  0.7071067812f,  0.3826834324f,
     0.0f, -0.3826834324f, -0.7071067812f, -0.9238795325f,
    -1.0f, -0.9238795325f, -0.7071067812f, -0.3826834324f };

// ---------------------------------------------------------------------------
// Kernel A: per-(b,k) FDN transfer function via Sherman-Morrison.
//   M = diag(1 + g_j d_j) - u v^T,  u_i = (2/|a|^2) a_i,  v_j = a_j g_j d_j
//   fdn = post^T diag(d) M^{-1} pre
// Writes the FULL 16000-bin spectrum (conjugate symmetric), scaled by 1/N.
// ---------------------------------------------------------------------------
__global__ void fdn_spectrum_kernel(const float* __restrict__ ds,
                                    const float* __restrict__ fb_adm,
                                    const float* __restrict__ fb_gain,
                                    const float* __restrict__ fb_filter,
                                    const float* __restrict__ pre_gain,
                                    const float* __restrict__ post_gain,
                                    float* __restrict__ Sre,
                                    float* __restrict__ Sim)
{
    int tid = blockIdx.x * blockDim.x + threadIdx.x;
    if (tid >= BATCH * NUM_FREQ) return;
    int b = tid / NUM_FREQ;
    int k = tid - b * NUM_FREQ;

    const float* a    = fb_adm    + b * NUM_CHANNEL;
    const float* gg   = fb_gain   + b * NUM_CHANNEL;
    const float* dd   = ds        + b * NUM_CHANNEL;
    const float* pre  = pre_gain  + b * NUM_CHANNEL;
    const float* post = post_gain + b * NUM_CHANNEL;
    const float* flt  = fb_filter + ((size_t)b * NUM_FREQ + (size_t)k) * NUM_CHANNEL;

    float asum = 0.f;
#pragma unroll
    for (int c = 0; c < NUM_CHANNEL; ++c) asum += a[c] * a[c];
    const float beta2 = 2.0f / asum;

    // accumulators (complex): s1 = v^T A^-1 pre ; s2 = v^T A^-1 u
    // tp = post^T d A^-1 pre ; tu = post^T d A^-1 u
    float s1r = 0.f, s1i = 0.f, s2r = 0.f, s2i = 0.f;
    float tpr = 0.f, tpi = 0.f, tur = 0.f, tui = 0.f;

#pragma unroll
    for (int c = 0; c < NUM_CHANNEL; ++c) {
        // delay d = exp(-i * 2*pi*k*ds/16000), range reduction in double
        double prod = (double)k * (double)dd[c];
        float  frac = (float)fmod(prod, (double)NUM_SAMPLE);
        float  phi  = -(TWO_PI / (float)NUM_SAMPLE) * frac;
        float  dre, dim_;
        __sincosf(phi, &dim_, &dre);           // dre = cos, dim_ = sin

        float g   = gg[c] * flt[c];            // fb_gain * fb_filter (real)
        // A_c = 1 + g*d  (complex), invA = conj(A)/|A|^2
        float Ar  = 1.0f + g * dre;
        float Ai  = g * dim_;
        float den = 1.0f / (Ar * Ar + Ai * Ai);
        float iAr =  Ar * den;
        float iAi = -Ai * den;

        float uc  = beta2 * a[c];
        float vr  = a[c] * g * dre;            // v_c = a*g*d (complex)
        float vi  = a[c] * g * dim_;

        // w = v * invA (complex)
        float wr = vr * iAr - vi * iAi;
        float wi = vr * iAi + vi * iAr;
        s1r += wr * pre[c];  s1i += wi * pre[c];
        s2r += wr * uc;      s2i += wi * uc;

        // q = post * d * invA (complex)
        float pdre = post[c] * dre, pdim = post[c] * dim_;
        float qr = pdre * iAr - pdim * iAi;
        float qi = pdre * iAi + pdim * iAr;
        tpr += qr * pre[c];  tpi += qi * pre[c];
        tur += qr * uc;      tui += qi * uc;
    }

    // fdn = tp + tu * s1 / (1 - s2)
    float er = 1.0f - s2r, ei = -s2i;
    float ed = 1.0f / (er * er + ei * ei);
    float nr = s1r * er + s1i * ei;            // s1 * conj(1-s2)
    float ni = s1i * er - s1r * ei;
    nr *= ed; ni *= ed;
    float fr = tpr + tur * nr - tui * ni;
    float fi = tpi + tur * ni + tui * nr;

    const float scale = 1.0f / (float)NUM_SAMPLE;   // irfft 1/N
    fr *= scale; fi *= scale;

    size_t base = (size_t)b * NUM_SAMPLE;
    Sre[base + k] = fr;
    Sim[base + k] = fi;
    if (k > 0 && k < NUM_SAMPLE / 2) {              // conjugate mirror
        Sre[base + (NUM_SAMPLE - k)] =  fr;
        Sim[base + (NUM_SAMPLE - k)] = -fi;
    }
}

// ---------------------------------------------------------------------------
// Kernel B: radix-16 stage.  k = k1 + 1000*k2, n = 16*m + r.
//   Y[b,k1,r] = sum_{k2} S[b, k1+1000*k2] * exp(+2*pi*i*k2*r/16)
//   Z = Y * exp(+2*pi*i*k1*r/16000)
// Z stored transposed [k1][j], j = b*16 + r  (coalesced B-operand for GEMM).
// k1 in [1000,1008) zero-padded.
// ---------------------------------------------------------------------------
__global__ void radix16_twiddle_kernel(const float* __restrict__ Sre,
                                       const float* __restrict__ Sim,
                                       float* __restrict__ Zre,
                                       float* __restrict__ Zim)
{
    int tid = blockIdx.x * blockDim.x + threadIdx.x;
    if (tid >= N1P * NJ) return;
    int k1 = tid >> 9;          // / 512
    int j  = tid & (NJ - 1);
    if (k1 >= N1) { Zre[tid] = 0.f; Zim[tid] = 0.f; return; }
    int b = j >> 4;
    int r = j & 15;

    const float* sr = Sre + (size_t)b * NUM_SAMPLE;
    const float* si = Sim + (size_t)b * NUM_SAMPLE;

    float yr = 0.f, yi = 0.f;
#pragma unroll
    for (int k2 = 0; k2 < N2; ++k2) {
        float xr = sr[k1 + N1 * k2];
        float xi = si[k1 + N1 * k2];
        int   p  = (k2 * r) & 15;                   // exact mod 16
        float c  = COS16[p];
        float s  = SIN16[p];
        yr += xr * c - xi * s;
        yi += xr * s + xi * c;
    }
    int   p2 = (k1 * r) % NUM_SAMPLE;               // exact int mod
    float c2, s2;
    __sincosf((TWO_PI / (float)NUM_SAMPLE) * (float)p2, &s2, &c2);
    Zre[tid] = yr * c2 - yi * s2;
    Zim[tid] = yr * s2 + yi * c2;
}

// ---------------------------------------------------------------------------
// Kernel C: 1000-point DFT stage as a WMMA f32 GEMM + FIR add.
//   out[m, j] = sum_{k1} cos(2*pi*m*k1/1000)*Zre[k1,j] - sin(.)*Zim[k1,j]
// One wave32 per 16x16 output tile (EXEC all ones, as WMMA requires).
// A-operand twiddles via complex-rotation recurrence (angle step is the
// per-lane constant 4*m*theta), resynced every 12 K-steps from an exact
// integer-mod sincos -> TRANS ops drop from 1008 to 43 per lane and the
// loop body becomes loads + FMA rotations + v_wmma.
// ---------------------------------------------------------------------------
__global__ __launch_bounds__(32)
void idft_wmma_kernel(const float* __restrict__ Zre,
                      const float* __restrict__ Zim,
                      const float* __restrict__ fir,
                      float* __restrict__ out)
{
    const int mtile = blockIdx.x;        // 63 tiles of 16 rows (m)
    const int b     = blockIdx.y;        // 32 batches (one 16-col j tile each)
    const int lane  = threadIdx.x;
    const int hi    = lane >> 4;         // 0 for lanes 0-15, 1 for 16-31
    const int lo    = lane & 15;

    const int m = mtile * 16 + lo;       // A-matrix row held by this lane
    const float th = TWO_PI / (float)N1;

    // per-lane rotation step e^{+i*4*m*theta} (exact integer range reduction)
    float C4, S4;
    __sincosf(th * (float)((4 * m) % N1), &S4, &C4);

    v8f acc = {0.f, 0.f, 0.f, 0.f, 0.f, 0.f, 0.f, 0.f};

    // B-operand pointers for this lane: column j = b*16 + lo, rows ka/kb
    const float* pr = Zre + (size_t)(2 * hi) * NJ + (size_t)b * 16 + lo;
    const float* pi_ = Zim + (size_t)(2 * hi) * NJ + (size_t)b * 16 + lo;

    // 252 K-steps of 4, in 21 resync blocks of 12
    for (int blk = 0; blk < N1P / 4; blk += 12) {
        // exact rotator state at block start: angles m*ka0, m*(ka0+1)
        int ka0 = 4 * blk + 2 * hi;
        float ca, sa, cb, sb;
        __sincosf(th * (float)((m * ka0) % N1), &sa, &ca);
        __sincosf(th * (float)((m * (ka0 + 1)) % N1), &sb, &cb);

#pragma unroll
        for (int t = 0; t < 12; ++t) {
            // prefetch Z 16 K-rows ahead (lowers to global_prefetch_b8)
            __builtin_prefetch(pr + (size_t)16 * NJ, 0, 1);
            __builtin_prefetch(pi_ + (size_t)16 * NJ, 0, 1);

            v2f aCos = { ca,  cb};
            v2f aSin = {-sa, -sb};
            v2f bRe  = { pr[0], pr[NJ] };
            v2f bIm  = { pi_[0], pi_[NJ] };

            acc = __builtin_amdgcn_wmma_f32_16x16x4_f32(
                      false, aCos, false, bRe, (short)0, acc, false, false);
            acc = __builtin_amdgcn_wmma_f32_16x16x4_f32(
                      false, aSin, false, bIm, (short)0, acc, false, false);

            // advance rotators by e^{+i*4*m*theta}
            float nca = ca * C4 - sa * S4;
            float nsa = ca * S4 + sa * C4;
            float ncb = cb * C4 - sb * S4;
            float nsb = cb * S4 + sb * C4;
            ca = nca; sa = nsa; cb = ncb; sb = nsb;

            pr  += 4 * NJ;                 // next K group of 4
            pi_ += 4 * NJ;
        }
    }

    // D layout: VGPR v holds row M = v + 8*hi, column N = lane&15.
    const int r = lo;                      // output column -> sample phase
#pragma unroll
    for (int v = 0; v < 8; ++v) {
        int mrow = mtile * 16 + v + 8 * hi;
        if (mrow < N1) {
            int   n   = 16 * mrow + r;     // n = 16*m + r
            float val = acc[v];
            if (n < 1024) val += fir[b * 1024 + n];
            out[(size_t)b * NUM_SAMPLE + n] = val;
        }
    }
}

// ---------------------------------------------------------------------------
// Launch
// ---------------------------------------------------------------------------
extern "C" void kernel_launch(void* const* d_in, const int* in_sizes, int n_in,
                              void* d_out, int out_size, void* d_ws, size_t ws_size,
                              hipStream_t stream)
{
    const float* ds        = (const float*)d_in[0];
    const float* fb_adm    = (const float*)d_in[1];
    const float* fb_gain   = (const float*)d_in[2];
    const float* fb_filter = (const float*)d_in[3];
    const float* pre_gain  = (const float*)d_in[4];
    const float* post_gain = (const float*)d_in[5];
    const float* fir       = (const float*)d_in[6];
    float* out = (float*)d_out;

    // workspace layout (floats):
    //   Sre[32*16000] | Sim[32*16000] | Zre[1008*512] | Zim[1008*512]
    //   total ~8.3 MB
    float* ws  = (float*)d_ws;
    float* Sre = ws;
    float* Sim = Sre + (size_t)BATCH * NUM_SAMPLE;
    float* Zre = Sim + (size_t)BATCH * NUM_SAMPLE;
    float* Zim = Zre + (size_t)N1P * NJ;

    {   // A: 32*8001 bins, one thread each
        int total  = BATCH * NUM_FREQ;
        int blocks = (total + 255) / 256;
        fdn_spectrum_kernel<<<blocks, 256, 0, stream>>>(
            ds, fb_adm, fb_gain, fb_filter, pre_gain, post_gain, Sre, Sim);
    }
    {   // B: 1008*512 Z entries, one thread each
        int total  = N1P * NJ;
        int blocks = (total + 255) / 256;
        radix16_twiddle_kernel<<<blocks, 256, 0, stream>>>(Sre, Sim, Zre, Zim);
    }
    {   // C: 63 m-tiles x 32 batches, one wave32 per 16x16 tile
        dim3 grid(63, 32);
        idft_wmma_kernel<<<grid, 32, 0, stream>>>(Zre, Zim, fir, out);
    }
}